// GraphLoss_13692355740435
// MI455X (gfx1250) — compile-verified
//
#include <hip/hip_runtime.h>
#include <cmath>

// Problem constants (match reference setup)
constexpr int BATCH = 16;
constexpr int N     = 1024;   // positions per sample (== G)
constexpr int ROW   = 1025;   // N + 1

typedef float v2f __attribute__((ext_vector_type(2)));
typedef float v8f __attribute__((ext_vector_type(8)));

// ---------------------------------------------------------------------------
// Kernel 1: zero the entire output buffer (dominant bandwidth cost: ~67 MB).
// float4 body + scalar tail (d_out is allocator-aligned, element 0 is 16B ok).
// ---------------------------------------------------------------------------
__global__ void gl_zero_kernel(float* __restrict__ out, long total) {
    const long nv     = total >> 2;
    float4*    o4     = (float4*)out;
    const long i      = (long)blockIdx.x * blockDim.x + threadIdx.x;
    const long stride = (long)gridDim.x * blockDim.x;
    const float4 z = make_float4(0.f, 0.f, 0.f, 0.f);
    for (long k = i; k < nv; k += stride) o4[k] = z;
    for (long k = (nv << 2) + i; k < total; k += stride) out[k] = 0.0f;
}

// Block-wide sum over 1024 threads (wave32). Result valid on thread 0.
__device__ inline float gl_block_sum(float v, float* red) {
    int tid  = threadIdx.x;
    int lane = tid & 31;
    int wid  = tid >> 5;
#pragma unroll
    for (int o = 16; o > 0; o >>= 1) v += __shfl_down(v, o, 32);
    if (lane == 0) red[wid] = v;
    __syncthreads();
    float r = 0.0f;
    if (wid == 0) {
        r = red[lane];
#pragma unroll
        for (int o = 16; o > 0; o >>= 1) r += __shfl_down(r, o, 32);
    }
    __syncthreads();
    return r;
}

// ---------------------------------------------------------------------------
// Kernel 2: one block (1024 threads = 32 waves) per batch sample.
//   Phase 1: nearest-neighbor via V_WMMA_F32_16X16X4_F32 (cross terms p.q).
//   Phase 2: instance/order keys, semantics one-hot, closs/semloss partials.
//   Phase 3: bitonic lexsort of packed keys in LDS.
//   Phase 4: consecutive-same-instance linking -> t_fwd / t_back.
//   Phase 5: scatter ones into matches_gt, gather match log-probs, reduce.
// ---------------------------------------------------------------------------
__global__ __launch_bounds__(1024) void gl_main_kernel(
    const float* __restrict__ matches,    // (B, 1025, 1025)
    const float* __restrict__ positions,  // (B, N, 2)
    const float* __restrict__ semantics,  // (B, 3, N)
    const float* __restrict__ gt_pts,     // (B, N, 2)
    const int*   __restrict__ gt_ins,     // (B, N)
    const int*   __restrict__ gt_order,   // (B, N)
    const int*   __restrict__ gt_type,    // (B, N)
    float*       __restrict__ out,        // [3 scalars][B*1025*1025][B*3*N]
    float*       __restrict__ ws,         // per-sample partial sums (B*3)
    float boundx, float boundy, float thr)
{
    __shared__ float    s_px[N], s_py[N];   // normalized positions
    __shared__ float    s_q2[2 * N];        // normalized gt, interleaved x,y
    __shared__ float    s_qn[N];            // |q|^2
    __shared__ float    s_nd2[N];           // min squared distance
    __shared__ int      s_near[N];          // argmin gt index
    __shared__ int      s_tf[N], s_tb[N];   // successor / predecessor
    __shared__ unsigned s_key[N];           // lexsort keys
    __shared__ float    s_red[32];

    const int b   = blockIdx.x;
    const int tid = threadIdx.x;

    // ---- Phase 0: cooperative load + normalize -----------------------------
    {
        float px = positions[((size_t)b * N + tid) * 2 + 0] / 399.0f; // NX-1
        float py = positions[((size_t)b * N + tid) * 2 + 1] / 199.0f;
        s_px[tid] = px;  s_py[tid] = py;
        float gx = (gt_pts[((size_t)b * N + tid) * 2 + 0] + boundx) / (2.0f * boundx);
        float gy = (gt_pts[((size_t)b * N + tid) * 2 + 1] + boundy) / (2.0f * boundy);
        s_q2[2 * tid + 0] = gx;
        s_q2[2 * tid + 1] = gy;
        s_qn[tid] = gx * gx + gy * gy;
        s_tf[tid] = N;   s_tb[tid] = N;
    }
    __syncthreads();

    // ---- Phase 1: WMMA nearest neighbor ------------------------------------
    // D = A x B with A = gt tile (16x4: K0=x, K1=y, K2=K3=0), B = pos tile
    // (4x16). C/D layout: lane owns column i = i_base + (lane&15); VGPR v
    // holds row j = j_base + v + (lane>=16 ? 8 : 0).  argmin over j is
    // lane-local; merge halves with one shfl_xor(16).
    // NOTE: LDS loads are unconditional (index in-range for all lanes); the
    // K=2/3 zeroing for lanes >= 16 is a value select (v_cndmask), so there is
    // no EXEC divergence anywhere near the WMMA.
    {
        const int  lane = tid & 31;
        const int  wave = tid >> 5;
        const bool lo   = lane < 16;
        const int  il   = lane & 15;
#pragma unroll
        for (int s = 0; s < 2; ++s) {
            const int ibase = (wave * 2 + s) * 16;
            const int i     = ibase + il;
            const float pxi = s_px[i], pyi = s_py[i];
            const float pn  = pxi * pxi + pyi * pyi;
            v2f bmat;
            bmat.x = lo ? pxi : 0.0f;   // K=0 row of B (lanes 0-15)
            bmat.y = lo ? pyi : 0.0f;   // K=1 row of B
            float best  = 3.0e38f;
            int   bestj = 0;
            for (int jb = 0; jb < N; jb += 16) {
                // one aligned ds_load_b64, then value-select to zero K=2/3
                const v2f q = *(const v2f*)&s_q2[2 * (jb + il)];
                v2f amat;
                amat.x = lo ? q.x : 0.0f;   // A K=0 (lanes 0-15, M=lane)
                amat.y = lo ? q.y : 0.0f;   // A K=1
                v8f cz = {0.f, 0.f, 0.f, 0.f, 0.f, 0.f, 0.f, 0.f};
                v8f c = __builtin_amdgcn_wmma_f32_16x16x4_f32(
                    /*neg_a=*/false, amat, /*neg_b=*/false, bmat,
                    /*c_mod=*/(short)0, cz, /*reuse_a=*/false, /*reuse_b=*/false);
                const int joff = jb + (lo ? 0 : 8);
#pragma unroll
                for (int v = 0; v < 8; ++v) {
                    const int   j  = joff + v;
                    const float d2 = pn + s_qn[j] - 2.0f * c[v];
                    if (d2 < best) { best = d2; bestj = j; }
                }
            }
            // merge the two M-halves (same column i lives in lane and lane^16)
            float ob = __shfl_xor(best, 16, 32);
            int   oj = __shfl_xor(bestj, 16, 32);
            if (ob < best || (ob == best && oj < bestj)) { best = ob; bestj = oj; }
            if (lo) { s_near[i] = bestj; s_nd2[i] = best; }
        }
    }
    __syncthreads();

    // ---- Phase 2: keys, semantics one-hot, closs/semloss partials ----------
    float closs_acc, semloss_acc;
    {
        const int   i  = tid;
        const float d2 = fmaxf(s_nd2[i], 0.0f);
        const float nd = sqrtf(d2 + 1e-12f);
        const int   nr = s_near[i];
        const int   ins = (nd < thr) ? gt_ins[(size_t)b * N + nr] : -1;
        const int   ord = gt_order[(size_t)b * N + nr];
        s_key[i] = ((unsigned)(ins + 1) << 15) | ((unsigned)ord << 10) | (unsigned)i;
        const int cls = gt_type[(size_t)b * N + nr];
        // semantics_gt one-hot (rest stays zero from the zero kernel)
        out[3 + (size_t)BATCH * ROW * ROW + ((size_t)b * 3 + cls) * N + i] = 1.0f;
        semloss_acc = -semantics[((size_t)b * 3 + cls) * N + i];
        closs_acc   = fabsf(s_px[i] - s_q2[2 * nr + 0]) +
                      fabsf(s_py[i] - s_q2[2 * nr + 1]);
    }
    __syncthreads();

    // ---- Phase 3: bitonic sort of 1024 keys (ascending == lexsort order) ---
    for (int k = 2; k <= N; k <<= 1) {
        for (int j = k >> 1; j > 0; j >>= 1) {
            const int ixj = tid ^ j;
            if (ixj > tid) {
                const unsigned a = s_key[tid];
                const unsigned c = s_key[ixj];
                const bool up = ((tid & k) == 0);
                if ((a > c) == up) { s_key[tid] = c; s_key[ixj] = a; }
            }
            __syncthreads();
        }
    }

    // ---- Phase 4: link consecutive sorted points within an instance --------
    if (tid < N - 1) {
        const unsigned ka = s_key[tid], kb = s_key[tid + 1];
        if ((ka >> 15) == (kb >> 15) && (ka >> 15) != 0u) {   // same ins, ins>=0
            const int a  = (int)(ka & 1023);
            const int bb = (int)(kb & 1023);
            s_tf[a]  = bb;   // distinct a's / distinct b's -> no races
            s_tb[bb] = a;
        }
    }
    __syncthreads();

    // ---- Phase 5: scatter matches_gt ones, gather match log-probs ----------
    float mloss_acc;
    {
        const int    i     = tid;
        const int    f     = s_tf[i];
        const int    bk    = s_tb[i];
        const size_t mbase = 3 + (size_t)b * ROW * ROW;
        out[mbase + (size_t)i * ROW + f] = 1.0f;              // row i: succ or N
        if (bk == N) out[mbase + (size_t)N * ROW + i] = 1.0f; // row N: orphans
        const size_t inbase = (size_t)b * ROW * ROW + (size_t)i * ROW;
        mloss_acc = -(matches[inbase + f] + matches[inbase + bk]);
    }

    // ---- Reductions -> per-sample workspace (deterministic, no atomics) ----
    const float cs = gl_block_sum(closs_acc,   s_red);
    const float ms = gl_block_sum(mloss_acc,   s_red);
    const float ss = gl_block_sum(semloss_acc, s_red);
    if (tid == 0) {
        ws[b * 3 + 0] = cs;
        ws[b * 3 + 1] = ms;
        ws[b * 3 + 2] = ss;
    }
}

// ---------------------------------------------------------------------------
// Kernel 3: deterministic fixed-order final reduction of the 3 scalars.
// ---------------------------------------------------------------------------
__global__ void gl_finalize_kernel(const float* __restrict__ ws,
                                   float* __restrict__ out) {
    if (blockIdx.x == 0 && threadIdx.x == 0) {
        float c = 0.f, m = 0.f, s = 0.f;
        for (int b = 0; b < BATCH; ++b) {
            c += ws[b * 3 + 0];
            m += ws[b * 3 + 1];
            s += ws[b * 3 + 2];
        }
        out[0] = c / (2.0f * N * BATCH);   // mean over (N,2) then batch
        out[1] = m / ((float)N * BATCH);
        out[2] = s / ((float)N * BATCH);
    }
}

extern "C" void kernel_launch(void* const* d_in, const int* in_sizes, int n_in,
                              void* d_out, int out_size, void* d_ws, size_t ws_size,
                              hipStream_t stream) {
    (void)in_sizes; (void)n_in; (void)ws_size;
    const float* matches   = (const float*)d_in[0];
    const float* positions = (const float*)d_in[1];
    const float* semantics = (const float*)d_in[2];
    // d_in[3] == masks (all ones, unused by the reference math)
    const float* gt_pts    = (const float*)d_in[4];
    const int*   gt_ins    = (const int*)d_in[5];
    const int*   gt_order  = (const int*)d_in[6];
    const int*   gt_type   = (const int*)d_in[7];
    float* out = (float*)d_out;
    float* ws  = (float*)d_ws;

    // Replicate numpy float32 constant derivation:
    //   BX = f32(xmin + dx/2);  BOUND = f32(dx)/2 - BX;  THR = ||1.5/(2*BOUND)||
    const float bxf    = (float)(-30.0 + 0.15 / 2.0);
    const float byf    = (float)(-15.0 + 0.15 / 2.0);
    const float dxh    = 0.15f / 2.0f;
    const float boundx = dxh - bxf;   // ~30.0
    const float boundy = dxh - byf;   // ~15.0
    const float rx = 1.5f / (2.0f * boundx);
    const float ry = 1.5f / (2.0f * boundy);
    const float thr = (float)sqrt((double)rx * rx + (double)ry * ry);

    gl_zero_kernel<<<2048, 256, 0, stream>>>(out, (long)out_size);
    gl_main_kernel<<<BATCH, 1024, 0, stream>>>(matches, positions, semantics,
                                               gt_pts, gt_ins, gt_order, gt_type,
                                               out, ws, boundx, boundy, thr);
    gl_finalize_kernel<<<1, 32, 0, stream>>>(ws, out);
}